// GCN_24627342475671
// MI455X (gfx1250) — compile-verified
//
#include <hip/hip_runtime.h>

#define N_NODES 100000
#define N_EDGES 1600000
#define L_LAYERS 8
#define LSTRIDE 68  // LDS row stride in floats: 68 mod 64 banks = 4 -> conflict-free b64 lane reads

typedef float v2f __attribute__((ext_vector_type(2)));
typedef float v8f __attribute__((ext_vector_type(8)));

__device__ __forceinline__ float leakyf(float x, float s) { return x > 0.0f ? x : s * x; }
__device__ __forceinline__ float nan0(float x) { return x != x ? 0.0f : x; }

// ---------- node encoder layer 1: [N,7] @ Wn1[7,64] + bn1, leaky 0.2 ----------
__global__ __launch_bounds__(256) void enc1_kernel(const float* __restrict__ x,
                                                   const float* __restrict__ W,
                                                   const float* __restrict__ b,
                                                   float* __restrict__ out) {
  int t = blockIdx.x * 256 + threadIdx.x;
  if (t >= N_NODES * 64) return;
  int n = t >> 6, j = t & 63;
  float s = b[j];
#pragma unroll
  for (int k = 0; k < 7; ++k) s += nan0(x[n * 7 + k]) * W[k * 64 + j];
  out[t] = leakyf(s, 0.2f);
}

// ---------- fused [M,64] @ [64,64] GEMM via V_WMMA_F32_16X16X4_F32 ----------
// Block = 256 thr = 8 waves, covers 64 rows x 64 cols.
// Stage W transposed + pre-activated A tile in LDS; each wave: 2 row-tiles x 1 col-tile,
// B fragment reused across both WMMAs. 32 v_wmma per wave, inner loop is ds_load_b64-fed.
// PRE : A element a -> leaky(a + pre_b[k],  pre_s)  (applied once at staging)
// POST: C element c -> leaky(c + post_b[j], post_s) (applied at store)
template <bool PRE, bool POST>
__global__ __launch_bounds__(256) void gemm64_wmma_kernel(
    const float* __restrict__ A, const float* __restrict__ W,
    const float* __restrict__ pre_b, float pre_s,
    const float* __restrict__ post_b, float post_s,
    float* __restrict__ out, int M) {
  __shared__ float Ws[64 * LSTRIDE];  // transposed: Ws[j*LSTRIDE + k]
  __shared__ float As[64 * LSTRIDE];  // As[r*LSTRIDE + k], pre-activation applied

  const int t = threadIdx.x;
  const int row0 = blockIdx.x * 64;

  // ---- stage W (transposed so B fragments are contiguous pairs) ----
  {
    int k = t >> 2;               // 0..63 (W row = fan_in index)
    int j0 = (t & 3) * 16;        // 16 output cols per thread
    const float4* src = (const float4*)(W + k * 64 + j0);
#pragma unroll
    for (int i = 0; i < 4; ++i) {
      float4 w4 = src[i];
      int j = j0 + i * 4;
      Ws[(j + 0) * LSTRIDE + k] = w4.x;
      Ws[(j + 1) * LSTRIDE + k] = w4.y;
      Ws[(j + 2) * LSTRIDE + k] = w4.z;
      Ws[(j + 3) * LSTRIDE + k] = w4.w;
    }
  }
  // ---- stage A tile (64 rows), pre-activation applied exactly once ----
  {
    int r = t >> 2;               // local row 0..63
    int c0 = (t & 3) * 16;
    int grow = row0 + r;
    if (grow >= M) grow = M - 1;  // clamp: keeps loads valid, stores guarded below
    const float4* src = (const float4*)(A + (size_t)grow * 64 + c0);
    float* dst = As + r * LSTRIDE + c0;
#pragma unroll
    for (int i = 0; i < 4; ++i) {
      float4 v = src[i];
      if (PRE) {
        int c = c0 + i * 4;
        v.x = leakyf(v.x + pre_b[c + 0], pre_s);
        v.y = leakyf(v.y + pre_b[c + 1], pre_s);
        v.z = leakyf(v.z + pre_b[c + 2], pre_s);
        v.w = leakyf(v.w + pre_b[c + 3], pre_s);
      }
      ((float4*)dst)[i] = v;
    }
  }
  __syncthreads();

  // ---- compute ----
  const int wave = t >> 5;        // 0..7
  const int lane = t & 31;
  const int half = lane >> 4;     // 0/1
  const int lrow = lane & 15;
  const int nloc = (wave & 3) * 16 + lrow;   // output column 0..63
  const int rg = wave >> 2;                  // row-tile-pair group 0/1
  const float* a0p = As + (rg * 32 + lrow) * LSTRIDE;       // row tile 2*rg
  const float* a1p = a0p + 16 * LSTRIDE;                    // row tile 2*rg+1
  const float* bp  = Ws + nloc * LSTRIDE;

  v8f acc0 = {0.f, 0.f, 0.f, 0.f, 0.f, 0.f, 0.f, 0.f};
  v8f acc1 = {0.f, 0.f, 0.f, 0.f, 0.f, 0.f, 0.f, 0.f};
#pragma unroll
  for (int kk = 0; kk < 16; ++kk) {
    int k0 = kk * 4 + half * 2;   // this lane's two K indices (A/B f32 frag layout)
    v2f b  = *(const v2f*)(bp + k0);
    v2f a0 = *(const v2f*)(a0p + k0);
    v2f a1 = *(const v2f*)(a1p + k0);
    acc0 = __builtin_amdgcn_wmma_f32_16x16x4_f32(false, a0, false, b, (short)0,
                                                 acc0, false, false);
    acc1 = __builtin_amdgcn_wmma_f32_16x16x4_f32(false, a1, false, b, (short)0,
                                                 acc1, false, false);
  }

  // C/D layout: VGPR r, lanes 0-15 -> M=r, lanes 16-31 -> M=r+8; N = lane%16
  float pb = POST ? post_b[nloc] : 0.0f;
  int mbase = row0 + rg * 32;
#pragma unroll
  for (int r = 0; r < 8; ++r) {
    int m0 = mbase + r + half * 8;
    int m1 = m0 + 16;
    float v0 = acc0[r], v1 = acc1[r];
    if (POST) { v0 = leakyf(v0 + pb, post_s); v1 = leakyf(v1 + pb, post_s); }
    if (m0 < M) out[(size_t)m0 * 64 + nloc] = v0;
    if (m1 < M) out[(size_t)m1 * 64 + nloc] = v1;
  }
}

// ---------- edge encoder: leaky(leaky(ea@We1+be1,0.2)@We2+be2, 0.005) ----------
__global__ __launch_bounds__(256) void edge_enc_kernel(
    const float* __restrict__ ea, const float* __restrict__ W1,
    const float* __restrict__ b1, const float* __restrict__ W2,
    const float* __restrict__ b2, float* __restrict__ ew) {
  int e = blockIdx.x * 256 + threadIdx.x;
  if (e >= N_EDGES) return;
  float a0 = nan0(ea[e * 4 + 0]);
  float a1 = nan0(ea[e * 4 + 1]);
  float a2 = nan0(ea[e * 4 + 2]);
  float a3 = nan0(ea[e * 4 + 3]);
  float s = b2[0];
#pragma unroll
  for (int j = 0; j < 16; ++j) {
    float h = a0 * W1[j] + a1 * W1[16 + j] + a2 * W1[32 + j] + a3 * W1[48 + j] + b1[j];
    s += leakyf(h, 0.2f) * W2[j];
  }
  ew[e] = leakyf(s, 0.005f);
}

// ---------- degree / norm ----------
__global__ __launch_bounds__(256) void deg_init_kernel(float* __restrict__ deg) {
  int n = blockIdx.x * 256 + threadIdx.x;
  if (n < N_NODES) deg[n] = 1.0f;  // self loop weight
}
__global__ __launch_bounds__(256) void deg_acc_kernel(const int* __restrict__ col,
                                                      const float* __restrict__ ew,
                                                      float* __restrict__ deg) {
  int e = blockIdx.x * 256 + threadIdx.x;
  if (e >= N_EDGES) return;
  atomicAdd(&deg[col[e]], ew[e]);
}
__global__ __launch_bounds__(256) void dis_kernel(const float* __restrict__ deg,
                                                  float* __restrict__ dis,
                                                  float* __restrict__ snorm) {
  int n = blockIdx.x * 256 + threadIdx.x;
  if (n >= N_NODES) return;
  float d = rsqrtf(deg[n]);        // deg<=0 -> inf/NaN like deg**-0.5
  if (isinf(d)) d = 0.0f;          // reference zeroes inf only
  dis[n] = d;
  snorm[n] = d * d;                // self-loop norm: dis[n]*1*dis[n]
}
__global__ __launch_bounds__(256) void norm_kernel(const int* __restrict__ row,
                                                   const int* __restrict__ col,
                                                   const float* __restrict__ ew,
                                                   const float* __restrict__ dis,
                                                   float* __restrict__ nrm) {
  int e = blockIdx.x * 256 + threadIdx.x;
  if (e >= N_EDGES) return;
  nrm[e] = dis[row[e]] * ew[e] * dis[col[e]];
}

// ---------- agg = h * snorm (self-loop contribution, also zero-init) ----------
__global__ __launch_bounds__(256) void init_agg_kernel(const float* __restrict__ h,
                                                       const float* __restrict__ snorm,
                                                       float* __restrict__ agg) {
  int t = blockIdx.x * 256 + threadIdx.x;  // one float4 per thread
  if (t >= N_NODES * 16) return;
  float s = snorm[t >> 4];
  float4 v = ((const float4*)h)[t];
  v.x *= s; v.y *= s; v.z *= s; v.w *= s;
  ((float4*)agg)[t] = v;
}

// ---------- edge scatter: agg[col] += h[row] * norm ----------
__global__ __launch_bounds__(256) void scatter_kernel(const float* __restrict__ h,
                                                      const int* __restrict__ row,
                                                      const int* __restrict__ col,
                                                      const float* __restrict__ nrm,
                                                      float* __restrict__ agg) {
  int t = blockIdx.x * 256 + threadIdx.x;  // 16 threads per edge, 4 floats each
  int e = t >> 4;
  int part = t & 15;
  if (e >= N_EDGES) return;
  int r = row[e], c = col[e];
  float w = nrm[e];
  float4 hv = ((const float4*)(h + (size_t)r * 64))[part];
  float* dst = agg + (size_t)c * 64 + part * 4;
  atomicAdd(dst + 0, hv.x * w);
  atomicAdd(dst + 1, hv.y * w);
  atomicAdd(dst + 2, hv.z * w);
  atomicAdd(dst + 3, hv.w * w);
}

// ---------- readout: [N,64] @ Wr[64,4] + br ----------
__global__ __launch_bounds__(256) void readout_kernel(const float* __restrict__ p,
                                                      const float* __restrict__ Wr,
                                                      const float* __restrict__ br,
                                                      float* __restrict__ out) {
  int n = blockIdx.x * 256 + threadIdx.x;
  if (n >= N_NODES) return;
  float acc0 = br[0], acc1 = br[1], acc2 = br[2], acc3 = br[3];
  const float* pr = p + (size_t)n * 64;
#pragma unroll
  for (int k = 0; k < 64; ++k) {
    float v = pr[k];
    acc0 += v * Wr[k * 4 + 0];
    acc1 += v * Wr[k * 4 + 1];
    acc2 += v * Wr[k * 4 + 2];
    acc3 += v * Wr[k * 4 + 3];
  }
  out[n * 4 + 0] = acc0;
  out[n * 4 + 1] = acc1;
  out[n * 4 + 2] = acc2;
  out[n * 4 + 3] = acc3;
}

extern "C" void kernel_launch(void* const* d_in, const int* in_sizes, int n_in,
                              void* d_out, int out_size, void* d_ws, size_t ws_size,
                              hipStream_t stream) {
  const float* x   = (const float*)d_in[0];
  const int*   ei  = (const int*)d_in[1];
  const float* ea  = (const float*)d_in[2];
  const float* Wn1 = (const float*)d_in[3];
  const float* bn1 = (const float*)d_in[4];
  const float* Wn2 = (const float*)d_in[5];
  const float* bn2 = (const float*)d_in[6];
  const float* We1 = (const float*)d_in[7];
  const float* be1 = (const float*)d_in[8];
  const float* We2 = (const float*)d_in[9];
  const float* be2 = (const float*)d_in[10];
  const float* Wg  = (const float*)d_in[11];
  const float* bg  = (const float*)d_in[12];
  const float* Wp1 = (const float*)d_in[13];
  const float* bp1 = (const float*)d_in[14];
  const float* Wp2 = (const float*)d_in[15];
  const float* bp2 = (const float*)d_in[16];
  const float* Wr  = (const float*)d_in[17];
  const float* br  = (const float*)d_in[18];
  float* out = (float*)d_out;

  // workspace carve-up (floats)
  float* ws    = (float*)d_ws;
  float* buf0  = ws;                          // N*64  (node / ping)
  float* buf1  = buf0 + (size_t)N_NODES * 64; // N*64  (h)
  float* buf2  = buf1 + (size_t)N_NODES * 64; // N*64  (agg / pong)
  float* ew    = buf2 + (size_t)N_NODES * 64; // E
  float* nrm   = ew + N_EDGES;                // E
  float* deg   = nrm + N_EDGES;               // N
  float* dis   = deg + N_NODES;               // N
  float* snorm = dis + N_NODES;               // N

  const int* row = ei;            // edge_index[0]
  const int* col = ei + N_EDGES;  // edge_index[1]

  const int gElem64 = (N_NODES * 64 + 255) / 256;  // 25000
  const int gGemm   = (N_NODES + 63) / 64;         // 1563
  const int gEdge   = (N_EDGES + 255) / 256;       // 6250
  const int gNode   = (N_NODES + 255) / 256;       // 391
  const int gVec4   = (N_NODES * 16 + 255) / 256;  // 6250
  const int gScat   = (N_EDGES * 16 + 255) / 256;  // 100000

  // node encoder
  enc1_kernel<<<gElem64, 256, 0, stream>>>(x, Wn1, bn1, buf1);
  gemm64_wmma_kernel<false, true><<<gGemm, 256, 0, stream>>>(
      buf1, Wn2, nullptr, 0.0f, bn2, 0.2f, buf0, N_NODES);

  // edge encoder + gcn_norm
  edge_enc_kernel<<<gEdge, 256, 0, stream>>>(ea, We1, be1, We2, be2, ew);
  deg_init_kernel<<<gNode, 256, 0, stream>>>(deg);
  deg_acc_kernel<<<gEdge, 256, 0, stream>>>(col, ew, deg);
  dis_kernel<<<gNode, 256, 0, stream>>>(deg, dis, snorm);
  norm_kernel<<<gEdge, 256, 0, stream>>>(row, col, ew, dis, nrm);

  // 8 GCN layers; bias+leaky of layer i-1 fused into layer i's GEMM staging
  float* in  = buf0;
  float* agg = buf2;
  for (int i = 0; i < L_LAYERS; ++i) {
    if (i == 0) {
      gemm64_wmma_kernel<false, false><<<gGemm, 256, 0, stream>>>(
          in, Wg + (size_t)i * 64 * 64, nullptr, 0.0f, nullptr, 0.0f, buf1, N_NODES);
    } else {
      gemm64_wmma_kernel<true, false><<<gGemm, 256, 0, stream>>>(
          in, Wg + (size_t)i * 64 * 64, bg + (size_t)(i - 1) * 64, 0.2f,
          nullptr, 0.0f, buf1, N_NODES);
    }
    init_agg_kernel<<<gVec4, 256, 0, stream>>>(buf1, snorm, agg);
    scatter_kernel<<<gScat, 256, 0, stream>>>(buf1, row, col, nrm, agg);
    float* tmp = in; in = agg; agg = tmp;
  }

  // post MLP (final GCN bias+leaky fused as pre-activation) + readout
  gemm64_wmma_kernel<true, true><<<gGemm, 256, 0, stream>>>(
      in, Wp1, bg + (size_t)(L_LAYERS - 1) * 64, 0.2f, bp1, 0.2f, buf1, N_NODES);
  gemm64_wmma_kernel<false, true><<<gGemm, 256, 0, stream>>>(
      buf1, Wp2, nullptr, 0.0f, bp2, 0.2f, agg, N_NODES);
  readout_kernel<<<gNode, 256, 0, stream>>>(agg, Wr, br, out);
}